// LineNNUEv1_39797166964915
// MI455X (gfx1250) — compile-verified
//
#include <hip/hip_runtime.h>
#include <hip/hip_bf16.h>

// LineNNUEv1 forward for gfx1250 (MI455X). wave32; f32 WMMA 16x16x4 for all
// dense heads. Memory-bound: dominant cost is the 118MB embedding gather
// (table is L2-resident at 113MB < 192MB L2) -> prefetch rows early.

typedef float v2f __attribute__((ext_vector_type(2)));
typedef float v8f __attribute__((ext_vector_type(8)));

#define DEV __device__ __forceinline__

// ---- async global->LDS staging (CDNA5 ASYNCcnt path), guarded ----
#if defined(__gfx1250__) && __has_builtin(__builtin_amdgcn_global_load_async_to_lds_b128) && \
    __has_builtin(__builtin_amdgcn_s_wait_asynccnt)
#define USE_ASYNC_LDS 1
// builtin signature: void(__device__ v4i*, __shared__ v4i*, imm int, imm int)
typedef int v4i_gcc __attribute__((vector_size(16)));
typedef __attribute__((address_space(1))) v4i_gcc as1_v4i;
typedef __attribute__((address_space(3))) v4i_gcc as3_v4i;
DEV as1_v4i* to_global(const void* p) {
  return (as1_v4i*)(unsigned long long)p;
}
DEV as3_v4i* to_lds(const void* p) {
  // generic shared pointer: low 32 bits are the LDS byte offset
  return (as3_v4i*)(unsigned int)(unsigned long long)p;
}
#endif

// ---- fake_quant helpers (round = nearest-even like jnp.round; floor variants) ----
DEV float fqr(float x, float s, float lo, float hi) {
  float xs = fminf(fmaxf(x * s, lo), hi);
  return rintf(xs) / s;
}
DEV float fqf(float x, float s, float lo, float hi) {
  float xs = fminf(fmaxf(x * s, lo), hi);
  return floorf(xs) / s;
}
DEV float qw8(float w)  { return fqr(w, 128.f, -128.f, 127.f); }                 // int8 weight @128
DEV float qb32(float b) { return fqr(b, 16384.f, -2147483648.f, 2147483647.f); } // int32 bias @128*128
DEV float fq8f(float t) { return fqf(t, 128.f, -128.f, 127.f); }

// ---- f32 WMMA 16x16x4: D = A(16x4) * B(4x16) + C ----
DEV v8f wmma4(v2f a, v2f b, v8f c) {
  return __builtin_amdgcn_wmma_f32_16x16x4_f32(false, a, false, b, (short)0, c, false, false);
}

// =====================================================================
// Kernel 1: embedding gather + quant + dir-mean + relu + depthwise conv
//           + region sums. One workgroup per batch sample, all in LDS.
// =====================================================================
__global__ __launch_bounds__(256) void k_feature(
    const int* __restrict__ line_encoding, const float* __restrict__ emb,
    const float* __restrict__ dw_w, const float* __restrict__ dw_b,
    float* __restrict__ value_in,  // [B,320]: cols 0..63 = feature_sum
    float* __restrict__ regions,   // [B,9,64] quantized
    float* __restrict__ pin)       // [B,32,225] = feat_dw (policy conv input)
{
  __shared__ __align__(16) float f[64 * 225];      // feature map [c][p]
  __shared__ __align__(16) float dwres[32 * 225];  // depthwise conv result
  __shared__ float rsum[9 * 64];
  __shared__ int   lidx[4 * 225];

  const int b = blockIdx.x, t = threadIdx.x;

  // load indices; prefetch each 256B embedding row while staging indices
  for (int i = t; i < 900; i += 256) {
    const int row = line_encoding[b * 900 + i];
    lidx[i] = row;
    __builtin_prefetch(&emb[(long)row * 64], 0, 3);       // global_prefetch_b8
    __builtin_prefetch(&emb[(long)row * 64 + 32], 0, 3);  // second cacheline of row
  }
  __syncthreads();

  // gather + clip[-16,16] + fq(32,16) + mean over 4 dirs + relu
  {
    const int c = t & 63, pof = t >> 6;
    for (int p0 = 0; p0 < 225; p0 += 4) {
      int p = p0 + pof;
      if (p < 225) {
        float s = 0.f;
#pragma unroll
        for (int d = 0; d < 4; d++) {
          int row = lidx[d * 225 + p];
          float e = emb[row * 64 + c];
          e = fminf(fmaxf(e, -16.f), 16.f);
          e = rintf(e * 32.f) * (1.f / 32.f);   // fq(e,32,16): no clip needed (|e*32|<=512)
          s += e;
        }
        f[c * 225 + p] = fmaxf(s * 0.25f, 0.f);
      }
    }
  }
  __syncthreads();

  // depthwise 3x3 conv on c<32 (quant='pixel-dwconv-floor')
  {
    const int cc = t & 31, pr = t >> 5;
    float wq[9];
#pragma unroll
    for (int k = 0; k < 9; k++) wq[k] = fqr(dw_w[cc * 9 + k], 65536.f, -32768.f, 32767.f);
    const float bq = fqr(dw_b[cc], 128.f, -32768.f, 32767.f);
    for (int p = pr; p < 225; p += 8) {
      const int hy = p / 15, wx = p % 15;
      float acc = 0.f;
#pragma unroll
      for (int dy = -1; dy <= 1; dy++)
#pragma unroll
        for (int dx = -1; dx <= 1; dx++) {
          const int y = hy + dy, x = wx + dx;
          float xv = 0.f;
          if (y >= 0 && y < 15 && x >= 0 && x < 15) {
            float fv = f[cc * 225 + y * 15 + x];
            xv = rintf(fminf(fmaxf(fv * 512.f, -32768.f), 32767.f)) * (1.f / 128.f); // fq(f*4,128,16)
          }
          float prod = xv * wq[(dy + 1) * 3 + (dx + 1)];
          prod = floorf(fminf(fmaxf(prod * 128.f, -2147483648.f), 2147483647.f)) * (1.f / 128.f);
          acc += prod;
        }
      acc = fmaxf(acc + bq, 0.f);
      dwres[cc * 225 + p] = fqr(acc, 128.f, -32768.f, 32767.f); // feat_dw
    }
  }
  // direction channels c>=32: quantize in place (disjoint from conv reads)
  for (int i = t; i < 32 * 225; i += 256) {
    const int c2 = 32 + i / 225, p = i % 225;
    f[c2 * 225 + p] = fqr(f[c2 * 225 + p], 128.f, -32768.f, 32767.f);
  }
  __syncthreads();

  // overwrite f[c<32] with feat_dw so f holds the full "feature" tensor
  for (int i = t; i < 32 * 225; i += 256) f[i] = dwres[i];
  __syncthreads();

  // region raw sums (9 regions of 5x5 blocks; H=W=15 -> [0,5,10,15])
  for (int rc = t; rc < 576; rc += 256) {
    const int r = rc / 64, ch = rc % 64, ri = r / 3, rj = r % 3;
    float s = 0.f;
    for (int y = ri * 5; y < ri * 5 + 5; y++)
      for (int x = rj * 5; x < rj * 5 + 5; x++)
        s += f[ch * 225 + y * 15 + x];
    rsum[r * 64 + ch] = s;
  }
  __syncthreads();

  // feature_sum = floor(total/2)/128 ; regions = floor(raw*4)/128
  if (t < 64) {
    float s = 0.f;
#pragma unroll
    for (int r = 0; r < 9; r++) s += rsum[r * 64 + t];
    value_in[b * 320 + t] =
        floorf(fminf(fmaxf(s * 0.5f, -2147483648.f), 2147483647.f)) * (1.f / 128.f);
  }
  for (int rc = t; rc < 576; rc += 256) {
    float s = rsum[rc];
    regions[b * 576 + rc] =
        floorf(fminf(fmaxf(s * 4.f, -2147483648.f), 2147483647.f)) * (1.f / 128.f);
  }
  for (int i = t; i < 32 * 225; i += 256) pin[b * 7200 + i] = dwres[i];
}

// =====================================================================
// Kernel 2: policy FC head. h = relu(fsum@pw1^T+b); o = h@pw2^T+b.
// M=512 K=64 N=64 then K=64 N=528. WMMA f32 16x16x4.
// =====================================================================
__global__ __launch_bounds__(128) void k_policy_fc(
    const float* __restrict__ value_in,
    const float* __restrict__ pw1w, const float* __restrict__ pw1b,
    const float* __restrict__ pw2w, const float* __restrict__ pw2b,
    float* __restrict__ obuf) // [B,528], cols<512 stored with 16b floor quant
{
  __shared__ __align__(16) float h[16 * 68];
  const int row0 = blockIdx.x * 16;
  const int tid = threadIdx.x, wave = tid >> 5, lane = tid & 31;
  const int hf = lane >> 4, mn = lane & 15;
  {
    const int col0 = wave * 16;
    v8f acc = {};
    for (int k0 = 0; k0 < 64; k0 += 4) {
      const int kk = k0 + 2 * hf;
      v2f a = *(const v2f*)(value_in + (row0 + mn) * 320 + kk);
      const float* wp = pw1w + (col0 + mn) * 64 + kk;
      v2f bv; bv.x = qw8(wp[0]); bv.y = qw8(wp[1]);
      acc = wmma4(a, bv, acc);
    }
    const float bias = qb32(pw1b[col0 + mn]);
#pragma unroll
    for (int v = 0; v < 8; v++)
      h[(v + 8 * hf) * 68 + col0 + mn] = fmaxf(acc[v] + bias, 0.f);
  }
  __syncthreads();
  for (int nt = wave; nt < 33; nt += 4) {
    const int col0 = nt * 16;
    v8f acc = {};
    for (int k0 = 0; k0 < 64; k0 += 4) {
      const int kk = k0 + 2 * hf;
      v2f a = *(const v2f*)(h + mn * 68 + kk);
      const float* wp = pw2w + (col0 + mn) * 64 + kk;
      v2f bv; bv.x = qw8(wp[0]); bv.y = qw8(wp[1]);
      acc = wmma4(a, bv, acc);
    }
    const float bias = qb32(pw2b[col0 + mn]);
#pragma unroll
    for (int v = 0; v < 8; v++) {
      const int row = row0 + v + 8 * hf, col = col0 + mn;
      float o = acc[v] + bias;
      if (col < 512) o = fqf(o, 16384.f, -32768.f, 32767.f); // fq(o,128*128,16,floor)
      obuf[row * 528 + col] = o;
    }
  }
}

// =====================================================================
// Kernel 3: per-sample dynamic pointwise conv + pout reduction -> policy
// pin staged via GLOBAL_LOAD_ASYNC_TO_LDS_B128 (ASYNCcnt) when available.
// =====================================================================
__global__ __launch_bounds__(256) void k_policy_conv(
    const float* __restrict__ pin, const float* __restrict__ obuf,
    const float* __restrict__ poutw, const float* __restrict__ poutb,
    float* __restrict__ out)
{
  __shared__ __align__(16) float pinS[32 * 225];
  __shared__ float pwS[16 * 32];
  __shared__ float pbS[16];
  __shared__ float poutS[16];
  __shared__ float pb0;
  const int b = blockIdx.x, t = threadIdx.x;
#if defined(USE_ASYNC_LDS)
  // 7200 floats = 1800 x 16B chunks; per-lane global->LDS async DMA
  for (int i = t; i < 1800; i += 256) {
    __builtin_amdgcn_global_load_async_to_lds_b128(
        to_global(pin + b * 7200 + i * 4), to_lds(pinS + i * 4), 0, 0);
  }
#else
  for (int i = t; i < 7200; i += 256) pinS[i] = pin[b * 7200 + i];
#endif
  for (int i = t; i < 512; i += 256)  pwS[i]  = obuf[b * 528 + i];
  if (t < 16) { pbS[t] = obuf[b * 528 + 512 + t]; poutS[t] = poutw[t]; }
  if (t == 0) pb0 = poutb[0];
#if defined(USE_ASYNC_LDS)
  __builtin_amdgcn_s_wait_asynccnt(0);
#endif
  __syncthreads();
  if (t < 225) {
    float psum = pb0;
#pragma unroll 4
    for (int i = 0; i < 16; i++) {
      float a = pbS[i];
      for (int c = 0; c < 32; c++) a += pinS[c * 225 + t] * pwS[i * 32 + c];
      psum += fmaxf(a, 0.f) * poutS[i];
    }
    out[1536 + b * 225 + t] = psum; // policy section after value [512,3]
  }
}

// =====================================================================
// Star block tile: 16 rows, N=64, K=64. x1/x2 WMMA -> relu*mul -> down WMMA.
// =====================================================================
DEV void star_tile(const float* A, int lda, int row0,
                   const float* w1, const float* b1,
                   const float* w2, const float* b2,
                   const float* w3, const float* b3,
                   float* sh, float* outt, int tid)
{
  const int wave = tid >> 5, lane = tid & 31, hf = lane >> 4, mn = lane & 15;
  const int col0 = wave * 16;
  v8f a1 = {}, a2 = {};
  for (int k0 = 0; k0 < 64; k0 += 4) {
    const int kk = k0 + 2 * hf;
    v2f av = *(const v2f*)(A + (row0 + mn) * lda + kk);
    const float* p1 = w1 + (col0 + mn) * 64 + kk;
    const float* p2 = w2 + (col0 + mn) * 64 + kk;
    v2f b1v; b1v.x = qw8(p1[0]); b1v.y = qw8(p1[1]);
    v2f b2v; b2v.x = qw8(p2[0]); b2v.y = qw8(p2[1]);
    a1 = wmma4(av, b1v, a1);
    a2 = wmma4(av, b2v, a2);
  }
  const float bb1 = qb32(b1[col0 + mn]), bb2 = qb32(b2[col0 + mn]);
#pragma unroll
  for (int v = 0; v < 8; v++) {
    float x1 = a1[v] + bb1, x2 = a2[v] + bb2;
    sh[(v + 8 * hf) * 68 + col0 + mn] = fmaxf(x1, 0.f) * x2;
  }
  __syncthreads();
  v8f a3 = {};
  for (int k0 = 0; k0 < 64; k0 += 4) {
    const int kk = k0 + 2 * hf;
    v2f av = *(const v2f*)(sh + mn * 68 + kk);
    const float* p3 = w3 + (col0 + mn) * 64 + kk;
    v2f b3v; b3v.x = qw8(p3[0]); b3v.y = qw8(p3[1]);
    a3 = wmma4(av, b3v, a3);
  }
  const float bb3 = qb32(b3[col0 + mn]);
#pragma unroll
  for (int v = 0; v < 8; v++) outt[(v + 8 * hf) * 68 + col0 + mn] = a3[v] + bb3;
  __syncthreads();
}

// Kernel 4: star over 9 regions (per-region weight set via idx map)
__global__ __launch_bounds__(128) void k_star1(
    const float* __restrict__ regions,
    const float* __restrict__ su1w, const float* __restrict__ su1b,
    const float* __restrict__ su2w, const float* __restrict__ su2b,
    const float* __restrict__ sdw,  const float* __restrict__ sdb,
    float* __restrict__ vbuf)
{
  __shared__ __align__(16) float sh[16 * 68];
  __shared__ __align__(16) float outt[16 * 68];
  const int idxmap[9] = {0, 1, 0, 1, 2, 1, 0, 1, 0};
  const int r = blockIdx.y, set = idxmap[r];
  const int row0 = blockIdx.x * 16;
  star_tile(regions + r * 64, 576, row0,
            su1w + set * 4096, su1b + set * 64,
            su2w + set * 4096, su2b + set * 64,
            sdw  + set * 4096, sdb  + set * 64,
            sh, outt, threadIdx.x);
  for (int i = threadIdx.x; i < 16 * 64; i += 128) {
    const int m = i >> 6, c = i & 63;
    vbuf[(row0 + m) * 576 + r * 64 + c] = outt[m * 68 + c];
  }
}

// Kernel 5: quadrant fq8-floor averaging tree
__global__ __launch_bounds__(256) void k_quads(
    const float* __restrict__ vbuf, float* __restrict__ quads)
{
  const int i = blockIdx.x * 256 + threadIdx.x;
  if (i >= 512 * 4 * 64) return;
  const int c = i & 63, q = (i >> 6) & 3, b = i >> 8;
  const int ra[4] = {0, 1, 3, 4}, rb[4] = {1, 2, 4, 5},
            rc[4] = {3, 4, 6, 7}, rd[4] = {4, 5, 7, 8};
  const float* base = vbuf + b * 576 + c;
  const float a_ = fq8f(base[ra[q] * 64]);
  const float b_ = fq8f(base[rb[q] * 64]);
  const float c_ = fq8f(base[rc[q] * 64]);
  const float d_ = fq8f(base[rd[q] * 64]);
  const float ab = fq8f((a_ + b_ + 0.0078125f) * 0.5f);
  const float cd = fq8f((c_ + d_ + 0.0078125f) * 0.5f);
  quads[i] = fq8f((ab + cd + 0.0078125f) * 0.5f);
}

// Kernel 6: star over quads (weight set 3), writes into value_in cols 64..319
__global__ __launch_bounds__(128) void k_star2(
    const float* __restrict__ quads,
    const float* __restrict__ su1w, const float* __restrict__ su1b,
    const float* __restrict__ su2w, const float* __restrict__ su2b,
    const float* __restrict__ sdw,  const float* __restrict__ sdb,
    float* __restrict__ value_in)
{
  __shared__ __align__(16) float sh[16 * 68];
  __shared__ __align__(16) float outt[16 * 68];
  const int row0 = blockIdx.x * 16;
  star_tile(quads, 64, row0,
            su1w + 3 * 4096, su1b + 3 * 64,
            su2w + 3 * 4096, su2b + 3 * 64,
            sdw  + 3 * 4096, sdb  + 3 * 64,
            sh, outt, threadIdx.x);
  for (int i = threadIdx.x; i < 16 * 64; i += 128) {
    const int m = i >> 6, c = i & 63, row = row0 + m;
    value_in[(row >> 2) * 320 + 64 + (row & 3) * 64 + c] = outt[m * 68 + c];
  }
}

// Kernel 7: value MLP 320->64->64->3
__global__ __launch_bounds__(128) void k_value(
    const float* __restrict__ value_in,
    const float* __restrict__ vl1w, const float* __restrict__ vl1b,
    const float* __restrict__ vl2w, const float* __restrict__ vl2b,
    const float* __restrict__ vl3w, const float* __restrict__ vl3b,
    float* __restrict__ out)
{
  __shared__ __align__(16) float h1[16 * 68];
  __shared__ __align__(16) float h2[16 * 68];
  const int row0 = blockIdx.x * 16;
  const int tid = threadIdx.x, wave = tid >> 5, lane = tid & 31;
  const int hf = lane >> 4, mn = lane & 15;
  const int col0 = wave * 16;
  {
    v8f acc = {};
    for (int k0 = 0; k0 < 320; k0 += 4) {
      const int kk = k0 + 2 * hf;
      v2f a = *(const v2f*)(value_in + (row0 + mn) * 320 + kk);
      const float* wp = vl1w + (col0 + mn) * 320 + kk;
      v2f bv; bv.x = qw8(wp[0]); bv.y = qw8(wp[1]);
      acc = wmma4(a, bv, acc);
    }
    const float bias = qb32(vl1b[col0 + mn]);
#pragma unroll
    for (int v = 0; v < 8; v++)
      h1[(v + 8 * hf) * 68 + col0 + mn] = fmaxf(acc[v] + bias, 0.f);
  }
  __syncthreads();
  {
    v8f acc = {};
    for (int k0 = 0; k0 < 64; k0 += 4) {
      const int kk = k0 + 2 * hf;
      v2f a = *(const v2f*)(h1 + mn * 68 + kk);
      const float* wp = vl2w + (col0 + mn) * 64 + kk;
      v2f bv; bv.x = qw8(wp[0]); bv.y = qw8(wp[1]);
      acc = wmma4(a, bv, acc);
    }
    const float bias = qb32(vl2b[col0 + mn]);
#pragma unroll
    for (int v = 0; v < 8; v++)
      h2[(v + 8 * hf) * 68 + col0 + mn] = fmaxf(acc[v] + bias, 0.f);
  }
  __syncthreads();
  if (tid < 48) {
    const int m = tid / 3, j = tid % 3;
    float s = qb32(vl3b[j]);
    for (int c = 0; c < 64; c++) s += h2[m * 68 + c] * qw8(vl3w[j * 64 + c]);
    out[(row0 + m) * 3 + j] = s;
  }
}

// =====================================================================
extern "C" void kernel_launch(void* const* d_in, const int* in_sizes, int n_in,
                              void* d_out, int out_size, void* d_ws, size_t ws_size,
                              hipStream_t stream) {
  (void)in_sizes; (void)n_in; (void)out_size; (void)ws_size;
  const int*   line  = (const int*)  d_in[0];
  const float* emb   = (const float*)d_in[1];
  const float* dww   = (const float*)d_in[2];
  const float* dwb   = (const float*)d_in[3];
  const float* pw1w  = (const float*)d_in[4];
  const float* pw1b  = (const float*)d_in[5];
  const float* pw2w  = (const float*)d_in[6];
  const float* pw2b  = (const float*)d_in[7];
  const float* poutw = (const float*)d_in[8];
  const float* poutb = (const float*)d_in[9];
  const float* su1w  = (const float*)d_in[10];
  const float* su1b  = (const float*)d_in[11];
  const float* su2w  = (const float*)d_in[12];
  const float* su2b  = (const float*)d_in[13];
  const float* sdw   = (const float*)d_in[14];
  const float* sdb   = (const float*)d_in[15];
  const float* vl1w  = (const float*)d_in[16];
  const float* vl1b  = (const float*)d_in[17];
  const float* vl2w  = (const float*)d_in[18];
  const float* vl2b  = (const float*)d_in[19];
  const float* vl3w  = (const float*)d_in[20];
  const float* vl3b  = (const float*)d_in[21];
  float* out = (float*)d_out;

  // workspace layout (floats): ~19.4 MB total
  float* ws       = (float*)d_ws;
  float* value_in = ws;            // [512,320]
  float* regions  = ws + 163840;   // [512,9,64]
  float* vbuf     = ws + 458752;   // [512,9,64]
  float* quadsb   = ws + 753664;   // [512*4,64]
  float* obuf     = ws + 884736;   // [512,528]
  float* pinb     = ws + 1155072;  // [512,32,225]

  k_feature<<<512, 256, 0, stream>>>(line, emb, dww, dwb, value_in, regions, pinb);
  k_policy_fc<<<32, 128, 0, stream>>>(value_in, pw1w, pw1b, pw2w, pw2b, obuf);
  k_policy_conv<<<512, 256, 0, stream>>>(pinb, obuf, poutw, poutb, out);
  k_star1<<<dim3(32, 9), 128, 0, stream>>>(regions, su1w, su1b, su2w, su2b, sdw, sdb, vbuf);
  k_quads<<<512, 256, 0, stream>>>(vbuf, quadsb);
  k_star2<<<128, 128, 0, stream>>>(quadsb, su1w, su1b, su2w, su2b, sdw, sdb, value_in);
  k_value<<<32, 128, 0, stream>>>(value_in, vl1w, vl1b, vl2w, vl2b, vl3w, vl3b, out);
}